// GCNModel_61443802137283
// MI455X (gfx1250) — compile-verified
//
#include <hip/hip_runtime.h>

typedef __attribute__((ext_vector_type(2))) float v2f;
typedef __attribute__((ext_vector_type(8))) float v8f;

#define Hdim 128
#define NEG_SLOPE 0.01f

// ---------------------------------------------------------------------------
// FP32 WMMA GEMM:  C[M,128] = act(A[M,K] @ B[K,128] + bias)
// One 16x16 output tile per wave, K-loop in steps of 4 using
// V_WMMA_F32_16X16X4_F32.  A is row-major [M,K], B row-major [K,128].
//
// A 16x4 layout : lanes 0-15 hold M=lane, VGPR0=K+0, VGPR1=K+1
//                 lanes 16-31 hold M=lane-16, VGPR0=K+2, VGPR1=K+3
// B 4x16 layout : lanes 0-15 hold N=lane, VGPR0=row K+0, VGPR1=row K+1
//                 lanes 16-31 hold N=lane-16, VGPR0=row K+2, VGPR1=row K+3
// C/D 16x16     : VGPR r -> row (half*8 + r), lane&15 -> col
// ---------------------------------------------------------------------------
template <bool LEAKY>
__global__ void gemm_wmma_f32(const float* __restrict__ A,
                              const float* __restrict__ B,
                              const float* __restrict__ bias,
                              float* __restrict__ C,
                              int M, int K) {
    const int wave   = (blockIdx.x * blockDim.x + threadIdx.x) >> 5;
    const int tilesM = M >> 4;              // M is a multiple of 16 (50000)
    const int tileM  = wave >> 3;           // 8 col-tiles (128/16)
    const int tileN  = wave & 7;
    if (tileM >= tilesM) return;            // uniform per wave -> EXEC all-ones

    const int lane = threadIdx.x & 31;
    const int half = lane >> 4;             // 0: lanes 0-15, 1: lanes 16-31
    const int lpos = lane & 15;

    const int rowA = tileM * 16 + lpos;     // M index this lane loads for A
    const int colB = tileN * 16 + lpos;     // N index this lane loads for B

    const float* __restrict__ Arow = A + (size_t)rowA * K;

    v8f acc = {};
    for (int k = 0; k < K; k += 4) {
        const int ka = k + half * 2;
        // A: two consecutive K values for this lane (8B aligned: ka is even)
        const float2 av = *(const float2*)(Arow + ka);
        v2f a; a.x = av.x; a.y = av.y;
        // B: rows ka, ka+1 at column colB
        v2f b;
        b.x = B[(size_t)ka * Hdim + colB];
        b.y = B[(size_t)(ka + 1) * Hdim + colB];
        acc = __builtin_amdgcn_wmma_f32_16x16x4_f32(
            /*neg_a=*/false, a, /*neg_b=*/false, b,
            /*c_mod=*/(short)0, acc, /*reuse_a=*/false, /*reuse_b=*/false);
    }

    const int col = tileN * 16 + lpos;
    const float bv = bias[col];
    #pragma unroll
    for (int r = 0; r < 8; ++r) {
        const int row = tileM * 16 + half * 8 + r;
        float v = acc[r] + bv;
        if (LEAKY) v = (v > 0.0f) ? v : v * NEG_SLOPE;
        C[(size_t)row * Hdim + col] = v;
    }
}

// ---------------------------------------------------------------------------
// Utility kernels
// ---------------------------------------------------------------------------
__global__ void zero_f4(float4* __restrict__ p, int n4) {
    int i = blockIdx.x * blockDim.x + threadIdx.x;
    if (i < n4) p[i] = make_float4(0.f, 0.f, 0.f, 0.f);
}

__global__ void copy_f4(float4* __restrict__ d, const float4* __restrict__ s, int n4) {
    int i = blockIdx.x * blockDim.x + threadIdx.x;
    if (i < n4) d[i] = s[i];
}

// out[dst[e]] += h[src[e]]          (32 lanes per edge, 4 floats per lane)
__global__ void scatter_plain(const float* __restrict__ h,
                              const int* __restrict__ src,
                              const int* __restrict__ dst,
                              float* __restrict__ out, int E) {
    int tid = blockIdx.x * blockDim.x + threadIdx.x;
    int e = tid >> 5;
    if (e >= E) return;
    int q = (tid & 31) << 2;
    int u = src[e], v = dst[e];
    const float4 val = *(const float4*)(h + (size_t)u * Hdim + q);
    float* o = out + (size_t)v * Hdim + q;
    atomicAdd(o + 0, val.x);
    atomicAdd(o + 1, val.y);
    atomicAdd(o + 2, val.z);
    atomicAdd(o + 3, val.w);
}

// out[dst[e]] += h[src[e]] / degree[src[e]]
__global__ void scatter_deg(const float* __restrict__ h,
                            const float* __restrict__ degree,
                            const int* __restrict__ src,
                            const int* __restrict__ dst,
                            float* __restrict__ out, int E) {
    int tid = blockIdx.x * blockDim.x + threadIdx.x;
    int e = tid >> 5;
    if (e >= E) return;
    int q = (tid & 31) << 2;
    int u = src[e], v = dst[e];
    const float inv = 1.0f / degree[u];
    const float4 val = *(const float4*)(h + (size_t)u * Hdim + q);
    float* o = out + (size_t)v * Hdim + q;
    atomicAdd(o + 0, val.x * inv);
    atomicAdd(o + 1, val.y * inv);
    atomicAdd(o + 2, val.z * inv);
    atomicAdd(o + 3, val.w * inv);
}

// colsum[f] += sum over a 64-row slab of h[row][f]   (128 threads/block)
__global__ void colsum_k(const float* __restrict__ h, float* __restrict__ out, int N) {
    int f = threadIdx.x;
    int row0 = blockIdx.x * 64;
    int rowEnd = row0 + 64; if (rowEnd > N) rowEnd = N;
    float acc = 0.f;
    for (int r = row0; r < rowEnd; ++r) acc += h[(size_t)r * Hdim + f];
    atomicAdd(&out[f], acc);
}

// Head: g = colsum/N ; p = g@Wpred+bpred ; logits = p@Wcls+bcls ; softmax
__global__ void head_k(const float* __restrict__ colsum,   // [4*128]
                       const float* __restrict__ Wpred,    // [512,128]
                       const float* __restrict__ bpred,    // [128]
                       const float* __restrict__ Wcls,     // [128,2]
                       const float* __restrict__ bcls,     // [2]
                       float* __restrict__ out, float invN) {
    __shared__ float g[512];
    __shared__ float p[128];
    int t = threadIdx.x;                    // 128 threads
    for (int i = t; i < 512; i += 128) g[i] = colsum[i] * invN;
    __syncthreads();
    float acc = bpred[t];
    for (int i = 0; i < 512; ++i) acc += g[i] * Wpred[(size_t)i * 128 + t];
    p[t] = acc;
    __syncthreads();
    if (t == 0) {
        float l0 = bcls[0], l1 = bcls[1];
        for (int j = 0; j < 128; ++j) {
            l0 += p[j] * Wcls[j * 2 + 0];
            l1 += p[j] * Wcls[j * 2 + 1];
        }
        float m  = fmaxf(l0, l1);
        float e0 = __expf(l0 - m), e1 = __expf(l1 - m);
        float s  = e0 + e1;
        out[0] = e0 / s;
        out[1] = e1 / s;
    }
}

// ---------------------------------------------------------------------------
// Host-side orchestration (graph-capture safe: kernels only)
// ---------------------------------------------------------------------------
extern "C" void kernel_launch(void* const* d_in, const int* in_sizes, int n_in,
                              void* d_out, int out_size, void* d_ws, size_t ws_size,
                              hipStream_t stream) {
    const float* node_feat = (const float*)d_in[0];
    const float* degree    = (const float*)d_in[3];
    const float* Wn        = (const float*)d_in[4];
    const float* bn        = (const float*)d_in[5];
    const float* Wgcn      = (const float*)d_in[8];
    const float* bgcn      = (const float*)d_in[9];
    const float* Wpred     = (const float*)d_in[10];
    const float* bpred     = (const float*)d_in[11];
    const float* Wcls      = (const float*)d_in[12];
    const float* bcls      = (const float*)d_in[13];
    const int*   src       = (const int*)d_in[14];
    const int*   dst       = (const int*)d_in[15];

    const int N  = in_sizes[3];             // degree: (N,1)
    const int E  = in_sizes[14];
    const int NF = in_sizes[0] / N;         // 256

    float* bufA    = (float*)d_ws;                    // [N,128]
    float* bufB    = bufA + (size_t)N * Hdim;         // [N,128]
    float* colsum  = bufB + (size_t)N * Hdim;         // [4*128]

    const int nh4   = (N * Hdim) / 4;
    const int zgrid = (nh4 + 255) / 256;
    const int sgrid = (E * 32 + 255) / 256;
    const int cgrid = (N + 63) / 64;
    const int gemm_grid = ((N >> 4) * 8 + 7) / 8;     // 8 waves/block of 256

    // zero the 4x128 column-sum block
    zero_f4<<<1, 256, 0, stream>>>((float4*)colsum, 128);

    // h_proj = node_feat @ Wn + bn        (WMMA fp32)
    gemm_wmma_f32<false><<<gemm_grid, 256, 0, stream>>>(node_feat, Wn, bn, bufA, N, NF);

    // stack slot 0: column sums of h_proj
    colsum_k<<<cgrid, 128, 0, stream>>>(bufA, colsum + 0 * Hdim, N);

    // fusion: h = segment_sum(h_proj[src], dst)   -> bufB
    zero_f4<<<zgrid, 256, 0, stream>>>((float4*)bufB, nh4);
    scatter_plain<<<sgrid, 256, 0, stream>>>(bufA, src, dst, bufB, E);

    // h lives in bufB; bufA is the aggregation scratch
    for (int l = 0; l < 3; ++l) {
        // agg = h  (self term)
        copy_f4<<<zgrid, 256, 0, stream>>>((float4*)bufA, (const float4*)bufB, nh4);
        // agg += segment_sum(h[src]/degree[src], dst)
        scatter_deg<<<sgrid, 256, 0, stream>>>(bufB, degree, src, dst, bufA, E);
        // h = leaky_relu(agg @ Wgcn + bgcn)       (WMMA fp32)
        gemm_wmma_f32<true><<<gemm_grid, 256, 0, stream>>>(bufA, Wgcn, bgcn, bufB, N, Hdim);
        // stack slot l+1
        colsum_k<<<cgrid, 128, 0, stream>>>(bufB, colsum + (l + 1) * Hdim, N);
    }

    // head: means -> Wpred -> Wcls -> softmax -> d_out[2]
    head_k<<<1, 128, 0, stream>>>(colsum, Wpred, bpred, Wcls, bcls,
                                  (float*)d_out, 1.0f / (float)N);
}